// Model_58969900974579
// MI455X (gfx1250) — compile-verified
//
#include <hip/hip_runtime.h>

typedef __attribute__((ext_vector_type(2))) float v2f;
typedef __attribute__((ext_vector_type(8))) float v8f;

#define BATCH    8
#define SEQ      64       // mamba sequence length = NVAR
#define ROWS     (BATCH*SEQ)   // 512
#define DMODEL   512
#define DSTATE   16
#define NPATCH   16
#define L_IN     48
#define NVAR     64
#define NSER     (BATCH*NVAR*NPATCH)  // 8192
#define PRED_LEN 96

__device__ __forceinline__ float apply_act(float v, int act) {
    if      (act == 1) v = fmaxf(v, 0.f);
    else if (act == 2) v = (v > 20.f) ? v : log1pf(expf(v));
    return v;
}

// ---------------------------------------------------------------------------
// fp32 GEMM on the WMMA pipe: C[M,N] = act(A[M,K] @ B[K,N] + bias)
// One wave32 computes a 32x32 output tile as 2x2 WMMA 16x16 tiles
// (V_WMMA_F32_16X16X4_F32, K-step 4). REQUIRES M%32==0, N%32==0, K%4==0 --
// true for every call site, so there are NO bounds checks. LDA/LDB are
// template constants so every load in the K-loop is an immediate-offset
// access off a single A pointer and a single B pointer (two u64 adds per
// iteration total), leaving the loop as loads + 4 WMMA chains.
// Fragment layouts per CDNA5 ISA 7.12.2:
//   A 16x4 : lanes 0-15 hold row M=lane  {K,K+1}; lanes 16-31 hold {K+2,K+3}
//   B 4x16 : lanes 0-15 hold col N=lane  {K,K+1}; lanes 16-31 hold {K+2,K+3}
//   C/D    : vgpr q -> row m0+q (lanes 0-15) / m0+8+q (lanes 16-31), col n0+(lane&15)
// act: 0=none 1=relu 2=softplus
// ---------------------------------------------------------------------------
template<int LDA, int LDB>
__global__ __launch_bounds__(256)
void gemm_f32_wmma(const float* __restrict__ A, const float* __restrict__ B,
                   float* __restrict__ C, int ldc,
                   int M, int N, int K,
                   const float* __restrict__ bias, int act)
{
    int wave = blockIdx.x * (blockDim.x >> 5) + (threadIdx.x >> 5);
    int lane = threadIdx.x & 31;
    int tnc = N >> 5;
    if (wave >= (M >> 5) * tnc) return;
    int tm = wave / tnc, tn = wave - tm * tnc;
    int m0 = tm << 5, n0 = tn << 5;
    int r   = lane & 15;
    bool hi = lane >= 16;
    int ka0 = hi ? 2 : 0;

    const float* pa = A + (m0 + r) * LDA + ka0;     // rows m0..m0+15 (+16*LDA: m0+16..31)
    const float* pb = B + ka0 * LDB + (n0 + r);     // cols n0..n0+15 (+16: n0+16..31)

    v8f acc00 = {0.f,0.f,0.f,0.f,0.f,0.f,0.f,0.f};
    v8f acc01 = acc00, acc10 = acc00, acc11 = acc00;

    #pragma unroll 4
    for (int k = 0; k < K; k += 4) {
        v2f a0 = *(const v2f*)pa;                   // global_load_b64
        v2f a1 = *(const v2f*)(pa + 16*LDA);        // global_load_b64 (imm offset)
        v2f b0, b1;
        b0.x = pb[0];        b0.y = pb[LDB];        // imm offsets off one base
        b1.x = pb[16];       b1.y = pb[LDB + 16];
        acc00 = __builtin_amdgcn_wmma_f32_16x16x4_f32(false, a0, false, b0, (short)0, acc00, false, false);
        acc01 = __builtin_amdgcn_wmma_f32_16x16x4_f32(false, a0, false, b1, (short)0, acc01, false, false);
        acc10 = __builtin_amdgcn_wmma_f32_16x16x4_f32(false, a1, false, b0, (short)0, acc10, false, false);
        acc11 = __builtin_amdgcn_wmma_f32_16x16x4_f32(false, a1, false, b1, (short)0, acc11, false, false);
        pa += 4;
        pb += 4 * LDB;
    }

    float bv0 = bias ? bias[n0 + r]      : 0.f;
    float bv1 = bias ? bias[n0 + 16 + r] : 0.f;
    int rb = m0 + (hi ? 8 : 0);
    float* c0 = C + (long)rb * ldc + n0 + r;
    #pragma unroll
    for (int q = 0; q < 8; ++q) {
        float* crow0 = c0 + q * ldc;            // row rb+q
        float* crow1 = crow0 + 16 * ldc;        // row rb+q+16
        crow0[0]  = apply_act(acc00[q] + bv0, act);
        crow0[16] = apply_act(acc01[q] + bv1, act);
        crow1[0]  = apply_act(acc10[q] + bv0, act);
        crow1[16] = apply_act(acc11[q] + bv1, act);
    }
}

// ---------------------------------------------------------------------------
// Fused TTCN: per series n (8192 total): build X(48x11) with time-embedding,
// two 31-wide relu MLPs, 48x31 @ 31x341 filter GEMM (WMMA in LDS, padded to
// 48x32 @ 32x352), masked softmax over L, einsum -> 31 features + mask flag.
// LDS budget ~125 KB (CDNA5: up to 320 KB per workgroup).
// ---------------------------------------------------------------------------
__global__ void ttcn_kernel(const float* __restrict__ x_enc,
                            const float* __restrict__ x_mark_enc,
                            const float* __restrict__ maskp,
                            const float* __restrict__ te_s_w, const float* __restrict__ te_s_b,
                            const float* __restrict__ te_p_w, const float* __restrict__ te_p_b,
                            const float* __restrict__ fg1_w, const float* __restrict__ fg1_b,
                            const float* __restrict__ fg2_w, const float* __restrict__ fg2_b,
                            const float* __restrict__ fg3_w, const float* __restrict__ fg3_b,
                            const float* __restrict__ t_bias,
                            float* __restrict__ xp)
{
    extern __shared__ float smem[];
    float* sB   = smem;              // 32*352  fg3 padded
    float* sF   = sB  + 32*352;      // 48*352  filt / softmax
    float* sX   = sF  + 48*352;      // 48*12   X padded
    float* sH1  = sX  + 48*12;       // 48*32
    float* sH2  = sH1 + 48*32;       // 48*32
    float* sM   = sH2 + 48*32;       // 48
    float* sRed = sM  + 48;          // 256

    const int tid = threadIdx.x;
    for (int idx = tid; idx < 32*352; idx += 256) {
        int k = idx / 352, j = idx - k*352;
        sB[idx] = (k < 31 && j < 341) ? fg3_w[k*341 + j] : 0.f;
    }
    const float tsw = te_s_w[0], tsb = te_s_b[0];

    for (int n = blockIdx.x; n < NSER; n += gridDim.x) {
        int b = n >> 10;
        int p = n & 15;
        int v = (n >> 4) & 63;
        int base = ((b*16 + p)*48)*64 + v;          // index of (b,p,l=0,v)

        if (tid < 48) sM[tid] = maskp[base + tid*64];
        for (int idx = tid; idx < 48*12; idx += 256) {
            int l = idx / 12, c = idx - l*12;
            float val = 0.f;
            if (c == 0) val = x_enc[base + l*64];
            else if (c < 11) {
                float t = x_mark_enc[base + l*64];
                val = (c == 1) ? (t*tsw + tsb)
                               : sinf(t*te_p_w[c-2] + te_p_b[c-2]);
            }
            sX[idx] = val;
        }
        __syncthreads();

        for (int idx = tid; idx < 48*32; idx += 256) {   // h1 = relu(X@fg1+b)
            int l = idx >> 5, k = idx & 31;
            float s = 0.f;
            if (k < 31) {
                s = fg1_b[k];
                for (int i = 0; i < 11; ++i) s += sX[l*12 + i]*fg1_w[i*31 + k];
                s = fmaxf(s, 0.f);
            }
            sH1[idx] = s;
        }
        __syncthreads();
        for (int idx = tid; idx < 48*32; idx += 256) {   // h2 = relu(h1@fg2+b)
            int l = idx >> 5, k = idx & 31;
            float s = 0.f;
            if (k < 31) {
                s = fg2_b[k];
                for (int i = 0; i < 31; ++i) s += sH1[l*32 + i]*fg2_w[i*31 + k];
                s = fmaxf(s, 0.f);
            }
            sH2[idx] = s;
        }
        __syncthreads();

        {   // filt = h2 @ fg3 + b, then mask; 3x22 tiles of 16x16 over 8 waves
            int wv = tid >> 5, lane = tid & 31;
            int r = lane & 15; bool hi = lane >= 16;
            for (int t5 = wv; t5 < 66; t5 += 8) {
                int tm = t5 / 22, tn = t5 - tm*22;
                int m0 = tm << 4, n0 = tn << 4;
                v8f acc = {0.f,0.f,0.f,0.f,0.f,0.f,0.f,0.f};
                #pragma unroll
                for (int kk = 0; kk < 32; kk += 4) {
                    int ka = kk + (hi ? 2 : 0);
                    v2f a, bb;
                    a.x  = sH2[(m0+r)*32 + ka];
                    a.y  = sH2[(m0+r)*32 + ka + 1];
                    bb.x = sB[ka*352 + n0 + r];
                    bb.y = sB[(ka+1)*352 + n0 + r];
                    acc = __builtin_amdgcn_wmma_f32_16x16x4_f32(false, a, false, bb,
                                                                (short)0, acc, false, false);
                }
                int j = n0 + r;
                if (j < 341) {
                    float fb = fg3_b[j];
                    int l0 = m0 + (hi ? 8 : 0);
                    #pragma unroll
                    for (int q = 0; q < 8; ++q) {
                        int l = l0 + q;
                        float mv = sM[l];
                        sF[l*352 + j] = (acc[q] + fb)*mv + (1.f - mv)*(-1e8f);
                    }
                }
            }
        }
        __syncthreads();

        for (int j = tid; j < 341; j += 256) {           // softmax over L
            float mx = -3.0e38f;
            for (int l = 0; l < 48; ++l) mx = fmaxf(mx, sF[l*352 + j]);
            float ssum = 0.f;
            for (int l = 0; l < 48; ++l) {
                float e = expf(sF[l*352 + j] - mx);
                sF[l*352 + j] = e; ssum += e;
            }
            float inv = 1.f / ssum;
            for (int l = 0; l < 48; ++l) sF[l*352 + j] *= inv;
        }
        __syncthreads();

        if (tid < 248) {   // out[k] = sum_{l,i} X[l,i]*fn[l,k*11+i], 8 partials/k
            int k = tid >> 3, seg = tid & 7;
            float s = 0.f;
            for (int l = seg; l < 48; l += 8)
                for (int i = 0; i < 11; ++i)
                    s += sX[l*12 + i]*sF[l*352 + k*11 + i];
            sRed[tid] = s;
        }
        if (tid == 248) {
            float ms = 0.f;
            for (int l = 0; l < 48; ++l) ms += sM[l];
            sRed[248] = (ms > 0.f) ? 1.f : 0.f;          // mpatch
        }
        __syncthreads();
        if (tid < 32) {
            float val;
            if (tid < 31) {
                float s = 0.f;
                #pragma unroll
                for (int q = 0; q < 8; ++q) s += sRed[tid*8 + q];
                val = fmaxf(s + t_bias[tid], 0.f);
            } else val = sRed[248];
            xp[(n >> 4)*512 + (n & 15)*32 + tid] = val;
        }
        __syncthreads();
    }
}

// ---------------------------------------------------------------------------
// Mamba pieces
// ---------------------------------------------------------------------------
__global__ void conv_silu_kernel(const float* __restrict__ xz,
                                 const float* __restrict__ conv_w,
                                 const float* __restrict__ conv_b,
                                 float* __restrict__ xc)
{
    int idx = blockIdx.x*blockDim.x + threadIdx.x;
    if (idx >= ROWS*DMODEL) return;
    int row = idx >> 9, d = idx & 511;
    int l = row & 63;
    float cur  = xz[row*1024 + d];
    float prev = (l > 0) ? xz[(row-1)*1024 + d] : 0.f;
    float v = prev*conv_w[d*2] + cur*conv_w[d*2+1] + conv_b[d];
    xc[idx] = v / (1.f + expf(-v));      // silu
}

__global__ void scan_kernel(const float* __restrict__ dtv,
                            const float* __restrict__ xc,
                            const float* __restrict__ dbl,
                            const float* __restrict__ xz,
                            const float* __restrict__ A_log,
                            const float* __restrict__ Dp,
                            float* __restrict__ ydash)
{
    int id = blockIdx.x*blockDim.x + threadIdx.x;
    if (id >= BATCH*DMODEL) return;
    int b = id >> 9, d = id & 511;
    float a[DSTATE], h[DSTATE];
    #pragma unroll
    for (int s = 0; s < DSTATE; ++s) { a[s] = -expf(A_log[d*DSTATE + s]); h[s] = 0.f; }
    float Dv = Dp[d];
    for (int l = 0; l < SEQ; ++l) {
        int row = b*SEQ + l;
        float dt_ = dtv[row*512 + d];
        float xv  = xc [row*512 + d];
        const float* dr = dbl + row*64;        // uniform per wave -> scalar loads
        float y = 0.f;
        #pragma unroll
        for (int s = 0; s < DSTATE; ++s) {
            float dA = expf(dt_*a[s]);
            h[s] = dA*h[s] + dt_*dr[32+s]*xv;
            y += h[s]*dr[48+s];
        }
        y += xv*Dv;
        float zv = xz[row*1024 + 512 + d];
        ydash[row*512 + d] = y * (zv / (1.f + expf(-zv)));
    }
}

__global__ void flip_kernel(const float* __restrict__ x, float* __restrict__ out)
{
    int idx = blockIdx.x*blockDim.x + threadIdx.x;
    if (idx >= ROWS*DMODEL) return;
    int d = idx & 511, row = idx >> 9, b = row >> 6, l = row & 63;
    out[(b*64 + (63 - l))*512 + d] = x[idx];
}

__global__ void add_flip_kernel(const float* __restrict__ x, const float* __restrict__ y1,
                                const float* __restrict__ y2, float* __restrict__ out)
{
    int idx = blockIdx.x*blockDim.x + threadIdx.x;
    if (idx >= ROWS*DMODEL) return;
    int d = idx & 511, row = idx >> 9, b = row >> 6, l = row & 63;
    out[idx] = x[idx] + y1[idx] + y2[(b*64 + (63 - l))*512 + d];
}

// LayerNorm over last dim (512): out = LN(a [+ bopt]) * g + beta
__global__ void ln_kernel(const float* __restrict__ a, const float* __restrict__ bopt,
                          const float* __restrict__ g, const float* __restrict__ beta,
                          float* __restrict__ out)
{
    __shared__ float red[256];
    int row = blockIdx.x, tid = threadIdx.x;
    int base = row*512;
    float v0 = a[base + tid], v1 = a[base + 256 + tid];
    if (bopt) { v0 += bopt[base + tid]; v1 += bopt[base + 256 + tid]; }
    red[tid] = v0 + v1;
    __syncthreads();
    for (int s = 128; s > 0; s >>= 1) { if (tid < s) red[tid] += red[tid+s]; __syncthreads(); }
    float mu = red[0] * (1.f/512.f);
    __syncthreads();
    float d0 = v0 - mu, d1 = v1 - mu;
    red[tid] = d0*d0 + d1*d1;
    __syncthreads();
    for (int s = 128; s > 0; s >>= 1) { if (tid < s) red[tid] += red[tid+s]; __syncthreads(); }
    float inv = rsqrtf(red[0]*(1.f/512.f) + 1e-5f);
    out[base + tid]       = d0*inv*g[tid]     + beta[tid];
    out[base + 256 + tid] = d1*inv*g[256+tid] + beta[256+tid];
}

// ---------------------------------------------------------------------------
// Decoder: exploit broadcast structure of h@dec1_w:
//   P[b,v,k] = enc[b,v,:] . W[:512,k]          (WMMA GEMM)
//   Q[b,t,k] = te(tt)     . W[512:,k] + b[k]   (tiny)
//   out[b,t,v] = relu(P+Q) . dec2_w + dec2_b
// ---------------------------------------------------------------------------
__global__ void dec_q_kernel(const float* __restrict__ x_mark_dec,
                             const float* __restrict__ te_s_w, const float* __restrict__ te_s_b,
                             const float* __restrict__ te_p_w, const float* __restrict__ te_p_b,
                             const float* __restrict__ dec1_w, const float* __restrict__ dec1_b,
                             float* __restrict__ Q)
{
    int idx = blockIdx.x*blockDim.x + threadIdx.x;
    if (idx >= BATCH*PRED_LEN*32) return;
    int k = idx & 31, bt = idx >> 5;
    float t = x_mark_dec[bt];
    float q = dec1_b[k] + (t*te_s_w[0] + te_s_b[0]) * dec1_w[512*32 + k];
    for (int j = 0; j < 9; ++j)
        q += sinf(t*te_p_w[j] + te_p_b[j]) * dec1_w[(513 + j)*32 + k];
    Q[idx] = q;
}

__global__ void dec_out_kernel(const float* __restrict__ P, const float* __restrict__ Q,
                               const float* __restrict__ dec2_w, const float* __restrict__ dec2_b,
                               float* __restrict__ out)
{
    int idx = blockIdx.x*blockDim.x + threadIdx.x;
    if (idx >= BATCH*PRED_LEN*NVAR) return;
    int v = idx & 63;
    int t = (idx >> 6) % PRED_LEN;
    int b = idx / (PRED_LEN*NVAR);
    const float* Pr = P + (b*64 + v)*32;
    const float* Qr = Q + (b*PRED_LEN + t)*32;
    float acc = dec2_b[0];
    #pragma unroll
    for (int k = 0; k < 32; ++k)
        acc += fmaxf(Pr[k] + Qr[k], 0.f) * dec2_w[k];
    out[idx] = acc;                                  // (b, t, v)
}

// ---------------------------------------------------------------------------
extern "C" void kernel_launch(void* const* d_in, const int* in_sizes, int n_in,
                              void* d_out, int out_size, void* d_ws, size_t ws_size,
                              hipStream_t stream)
{
    (void)in_sizes; (void)n_in; (void)out_size; (void)ws_size;
    auto F = [&](int i) { return (const float*)d_in[i]; };
    // setup_inputs order: x_enc, x_mark_enc, x_dec, x_mark_dec, mask, params(flattened)
    const float* x_enc      = F(0);
    const float* x_mark_enc = F(1);
    const float* x_mark_dec = F(3);
    const float* maskp      = F(4);
    const float* te_s_w = F(5),  *te_s_b = F(6);
    const float* te_p_w = F(7),  *te_p_b = F(8);
    const float* fg1_w  = F(9),  *fg1_b  = F(10);
    const float* fg2_w  = F(11), *fg2_b  = F(12);
    const float* fg3_w  = F(13), *fg3_b  = F(14);
    const float* t_bias = F(15);
    const float* emb_w  = F(16), *emb_b  = F(17);
    const int LBASE = 18, LSTRIDE = 26;   // per layer: 9(m_fwd)+9(m_bwd)+8
    const float* norm_g = F(LBASE + 2*LSTRIDE + 0);
    const float* norm_b = F(LBASE + 2*LSTRIDE + 1);
    const float* dec1_w = F(LBASE + 2*LSTRIDE + 2);
    const float* dec1_b = F(LBASE + 2*LSTRIDE + 3);
    const float* dec2_w = F(LBASE + 2*LSTRIDE + 4);
    const float* dec2_b = F(LBASE + 2*LSTRIDE + 5);

    float* w = (float*)d_ws;
    float* xp    = w; w += 262144;
    float* encA  = w; w += 262144;
    float* encB  = w; w += 262144;
    float* tmpS  = w; w += 262144;
    float* xz    = w; w += 524288;
    float* xc    = w; w += 262144;
    float* dbl   = w; w += 32768;
    float* dtbuf = w; w += 262144;
    float* ydash = w; w += 262144;
    float* y1b   = w; w += 262144;
    float* y2b   = w; w += 262144;
    float* xfb   = w; w += 262144;
    float* ffn1  = w; w += 262144;
    float* Pbuf  = w; w += 16384;
    float* Qbuf  = w; w += 24576;   // total ~13.8 MB

    // gemm<LDA,LDB> dispatch: one wave per 32x32 tile
    auto nblocks = [](int M, int N) { return (((M >> 5) * (N >> 5)) + 7) / 8; };

    auto run_mamba = [&](const float* xin, float* yout, int pb) {
        const float* in_proj = F(pb+0);
        const float* conv_w  = F(pb+1);
        const float* conv_b  = F(pb+2);
        const float* x_proj  = F(pb+3);
        const float* dt_w    = F(pb+4);
        const float* dt_b    = F(pb+5);
        const float* A_log   = F(pb+6);
        const float* Dp      = F(pb+7);
        const float* out_prj = F(pb+8);
        gemm_f32_wmma<512,1024><<<nblocks(ROWS,1024), 256, 0, stream>>>(
            xin, in_proj, xz, 1024, ROWS, 1024, 512, nullptr, 0);
        conv_silu_kernel<<<1024, 256, 0, stream>>>(xz, conv_w, conv_b, xc);
        gemm_f32_wmma<512,64><<<nblocks(ROWS,64), 256, 0, stream>>>(
            xc, x_proj, dbl, 64, ROWS, 64, 512, nullptr, 0);
        gemm_f32_wmma<64,512><<<nblocks(ROWS,512), 256, 0, stream>>>(
            dbl, dt_w, dtbuf, 512, ROWS, 512, 32, dt_b, 2);           // softplus
        scan_kernel<<<16, 256, 0, stream>>>(dtbuf, xc, dbl, xz, A_log, Dp, ydash);
        gemm_f32_wmma<512,512><<<nblocks(ROWS,512), 256, 0, stream>>>(
            ydash, out_prj, yout, 512, ROWS, 512, 512, nullptr, 0);
    };

    size_t smem_ttcn = (size_t)(32*352 + 48*352 + 48*12 + 48*32*2 + 48 + 256)*sizeof(float);
    ttcn_kernel<<<512, 256, smem_ttcn, stream>>>(x_enc, x_mark_enc, maskp,
        te_s_w, te_s_b, te_p_w, te_p_b, fg1_w, fg1_b, fg2_w, fg2_b,
        fg3_w, fg3_b, t_bias, xp);

    gemm_f32_wmma<512,512><<<nblocks(ROWS,512), 256, 0, stream>>>(
        xp, emb_w, encA, 512, ROWS, 512, 512, emb_b, 0);

    float* cur = encA; float* nxt = encB;
    for (int L = 0; L < 2; ++L) {
        int base = LBASE + L*LSTRIDE;
        run_mamba(cur, y1b, base + 0);
        flip_kernel<<<1024, 256, 0, stream>>>(cur, xfb);
        run_mamba(xfb, y2b, base + 9);
        add_flip_kernel<<<1024, 256, 0, stream>>>(cur, y1b, y2b, tmpS);
        ln_kernel<<<ROWS, 256, 0, stream>>>(tmpS, nullptr, F(base+22), F(base+23), nxt);
        gemm_f32_wmma<512,512><<<nblocks(ROWS,512), 256, 0, stream>>>(
            nxt, F(base+18), ffn1, 512, ROWS, 512, 512, F(base+19), 1);   // relu
        gemm_f32_wmma<512,512><<<nblocks(ROWS,512), 256, 0, stream>>>(
            ffn1, F(base+20), tmpS, 512, ROWS, 512, 512, F(base+21), 0);
        ln_kernel<<<ROWS, 256, 0, stream>>>(nxt, tmpS, F(base+24), F(base+25), cur);
    }
    ln_kernel<<<ROWS, 256, 0, stream>>>(cur, nullptr, norm_g, norm_b, nxt);

    gemm_f32_wmma<512,32><<<nblocks(ROWS,32), 256, 0, stream>>>(
        nxt, dec1_w, Pbuf, 32, ROWS, 32, 512, nullptr, 0);
    dec_q_kernel<<<(BATCH*PRED_LEN*32 + 255)/256, 256, 0, stream>>>(
        x_mark_dec, te_s_w, te_s_b, te_p_w, te_p_b, dec1_w, dec1_b, Qbuf);
    dec_out_kernel<<<(BATCH*PRED_LEN*NVAR + 255)/256, 256, 0, stream>>>(
        Pbuf, Qbuf, dec2_w, dec2_b, (float*)d_out);
}